// PINNLoss_RES_78881369358757
// MI455X (gfx1250) — compile-verified
//
#include <hip/hip_runtime.h>
#include <math.h>

// P = DT^2 * C^2 / DH^2 = (0.002^2 * 2200^2) / 5.0^2 = 0.7744
#define P_COEF 0.7744f

typedef float v2f __attribute__((ext_vector_type(2)));
typedef float v8f __attribute__((ext_vector_type(8)));

// One block = 2 rows (512 cols) of one image. 128 threads/row, 4 elems/thread.
__global__ __launch_bounds__(256) void pinn_residual_kernel(
    const float* __restrict__ fem, float* __restrict__ partials) {
  const int b   = blockIdx.x;       // 0 .. 64*256-1
  const int img = b >> 8;           // image index
  const int blk = b & 255;          // row-pair within image
  const int tid = threadIdx.x;
  const int y   = (blk << 1) + (tid >> 7);   // 0..511
  const int x   = (tid & 127) << 2;          // 0..508 step 4

  const float* __restrict__ base = fem + (size_t)img * (4u * 512u * 512u);
  const float* __restrict__ unm1 = base + 1u * 512u * 512u;  // channel -3
  const float* __restrict__ un   = base + 2u * 512u * 512u;  // channel -2
  const float* __restrict__ upr  = base + 3u * 512u * 512u;  // channel -1

  const int ym = (y == 0) ? 511 : (y - 1);
  const int yp = (y == 511) ? 0 : (y + 1);

  const float4 c  = *(const float4*)(un   + y  * 512 + x);  // u_n center
  const float4 u4 = *(const float4*)(un   + ym * 512 + x);  // u_n row above (wrap)
  const float4 d4 = *(const float4*)(un   + yp * 512 + x);  // u_n row below (wrap)
  const float4 m1 = *(const float4*)(unm1 + y  * 512 + x);  // u_nm1
  const float4 pr = *(const float4*)(upr  + y  * 512 + x);  // u_pred
  const float  lft = un[y * 512 + ((x == 0)   ? 511 : (x - 1))];
  const float  rgt = un[y * 512 + ((x == 508) ? 0   : (x + 4))];

  // r = P*(L+R+U+D) + (2-4P)*center - u_nm1 - u_pred
  const float k = 2.0f - 4.0f * P_COEF;
  const float r0 = P_COEF * (lft + c.y + u4.x + d4.x) + k * c.x - m1.x - pr.x;
  const float r1 = P_COEF * (c.x + c.z + u4.y + d4.y) + k * c.y - m1.y - pr.y;
  const float r2 = P_COEF * (c.y + c.w + u4.z + d4.z) + k * c.z - m1.z - pr.z;
  const float r3 = P_COEF * (c.z + rgt + u4.w + d4.w) + k * c.w - m1.w - pr.w;

  const float acc = r0 * r0 + r1 * r1 + r2 * r2 + r3 * r3;

  // ---- wave32 sum via the matrix pipe: D = A(16x4 data) x B(4x16 ones) ----
  // A layout (32-bit 16x4): lanes 0-15 -> (M=lane, K=0) in VGPR0; lanes 16-31
  // -> (M=lane-16, K=2) in VGPR0. Put acc in VGPR0 (A.x), zero in VGPR1.
  // Then D[m][n] = acc[m] + acc[m+16] for every n.
  v2f A; A.x = acc;  A.y = 0.0f;
  v2f B; B.x = 1.0f; B.y = 1.0f;   // B = all ones
  v8f Dm = {};
  Dm = __builtin_amdgcn_wmma_f32_16x16x4_f32(
      /*neg_a=*/false, A, /*neg_b=*/false, B,
      /*c_mod=*/(short)0, Dm, /*reuse_a=*/false, /*reuse_b=*/false);

  // Per-lane: s = sum over the 8 D VGPRs. Lanes 0-15 all hold
  // sum_{m=0..7}(acc[m]+acc[m+16]); lanes 16-31 hold sum_{m=8..15}(...).
  float s = Dm[0] + Dm[1] + Dm[2] + Dm[3] + Dm[4] + Dm[5] + Dm[6] + Dm[7];
  float wave_sum = __shfl(s, 0, 32) + __shfl(s, 16, 32);

  // 8 waves per block -> LDS -> single partial per block.
  __shared__ float sh[8];
  if ((tid & 31) == 0) sh[tid >> 5] = wave_sum;
  __syncthreads();
  if (tid == 0) {
    partials[b] = sh[0] + sh[1] + sh[2] + sh[3] + sh[4] + sh[5] + sh[6] + sh[7];
  }
}

// Deterministic fixed-order reduction of the 16384 block partials + sqrt.
__global__ __launch_bounds__(1024) void final_reduce_kernel(
    const float* __restrict__ partials, int n, float* __restrict__ out) {
  __shared__ float sh[1024];
  float s = 0.0f;
  for (int i = threadIdx.x; i < n; i += 1024) s += partials[i];
  sh[threadIdx.x] = s;
  __syncthreads();
  for (int off = 512; off > 0; off >>= 1) {
    if ((int)threadIdx.x < off) sh[threadIdx.x] += sh[threadIdx.x + off];
    __syncthreads();
  }
  if (threadIdx.x == 0) out[0] = sqrtf(sh[0]);
}

extern "C" void kernel_launch(void* const* d_in, const int* in_sizes, int n_in,
                              void* d_out, int out_size, void* d_ws, size_t ws_size,
                              hipStream_t stream) {
  const float* fem = (const float*)d_in[0];   // (64, 4, 512, 512) f32
  float* out      = (float*)d_out;            // scalar f32
  float* partials = (float*)d_ws;             // 16384 floats = 64 KB scratch

  const int nblocks = 64 * 256;               // 2 rows per block
  pinn_residual_kernel<<<nblocks, 256, 0, stream>>>(fem, partials);
  final_reduce_kernel<<<1, 1024, 0, stream>>>(partials, nblocks, out);
}